// KKTLoss_16819091931477
// MI455X (gfx1250) — compile-verified
//
#include <hip/hip_runtime.h>

// Problem constants (match reference).
#define BB   32
#define MM   16384
#define NN   16384
#define NNZ  524288
#define LOG_MN 14        // M == N == 2^14
// flattened element counts
#define BM   (BB * MM)   // 524288
#define BN   (BB * NN)   // 524288
#define TOTAL_NNZ (BB * NNZ)  // 16777216

typedef float v2f  __attribute__((ext_vector_type(2)));
typedef float v4f  __attribute__((ext_vector_type(4)));
typedef float v8f  __attribute__((ext_vector_type(8)));
typedef int   v4i  __attribute__((ext_vector_type(4)));

// ---------------------------------------------------------------------------
// Kernel 0: zero the two scatter accumulators (4 MB in d_ws) and d_out.
// ---------------------------------------------------------------------------
__global__ void kkt_zero(float* __restrict__ ax_acc,
                         float* __restrict__ at_acc,
                         float* __restrict__ out) {
  int tid = blockIdx.x * blockDim.x + threadIdx.x;
  int stride = gridDim.x * blockDim.x;
  for (int i = tid; i < BM; i += stride) {
    ax_acc[i] = 0.0f;
    at_acc[i] = 0.0f;
  }
  if (tid == 0) out[0] = 0.0f;
}

// ---------------------------------------------------------------------------
// Kernel 1: COO scatter. Streams 192 MB of vals/rows/cols with non-temporal
// 128-bit loads (keep L2 for the hot accumulators + x/lam), gathers x[c] and
// lam[r] (L2-resident, 4 MB), and scatter-adds with global_atomic_add_f32
// into the L2-resident accumulators.
// ---------------------------------------------------------------------------
__global__ void kkt_scatter(const float* __restrict__ x,     // [B*N]
                            const float* __restrict__ lam,   // [B*M]
                            const v4f*  __restrict__ vals,   // [B*NNZ/4]
                            const v4i*  __restrict__ rows,
                            const v4i*  __restrict__ cols,
                            float* __restrict__ ax_acc,      // [B*M]
                            float* __restrict__ at_acc) {    // [B*N]
  const int nvec = TOTAL_NNZ / 4;              // 4M vec4 groups
  int tid = blockIdx.x * blockDim.x + threadIdx.x;
  int stride = gridDim.x * blockDim.x;
  for (int i = tid; i < nvec; i += stride) {
    v4f v = __builtin_nontemporal_load(&vals[i]);
    v4i r = __builtin_nontemporal_load(&rows[i]);
    v4i c = __builtin_nontemporal_load(&cols[i]);
    // NNZ = 2^19 -> vec4 groups per problem = 2^17; group never straddles b.
    int b = i >> 17;
    long base = ((long)b) << LOG_MN;           // b*M == b*N
    const float* xb  = x      + base;
    const float* lb  = lam    + base;
    float*       axb = ax_acc + base;
    float*       atb = at_acc + base;

    atomicAdd(&axb[r.x], v.x * xb[c.x]);
    atomicAdd(&atb[c.x], v.x * lb[r.x]);
    atomicAdd(&axb[r.y], v.y * xb[c.y]);
    atomicAdd(&atb[c.y], v.y * lb[r.y]);
    atomicAdd(&axb[r.z], v.z * xb[c.z]);
    atomicAdd(&atb[c.z], v.z * lb[r.z]);
    atomicAdd(&axb[r.w], v.w * xb[c.w]);
    atomicAdd(&atb[c.w], v.w * lb[r.w]);
  }
}

// ---------------------------------------------------------------------------
// Exact f32 wave-wide sum via V_WMMA_F32_16X16X4_F32.
// A (16x4) carries one partial per lane in its first A-VGPR (second = 0);
// B (4x16) = all ones => every column n of D sums ALL A entries:
//   D[m][n] = sum_k A[m][k].  Column n lives in lanes {n, n+16}, 8 VGPRs each,
// so (sum of 8 C registers) + shfl_xor(16) = grand total in every lane.
// Products are x*1.0 (exact), accumulation is f32: bit-exactness preserved.
// Requires EXEC all-ones: callers are divergence-free full waves.
// ---------------------------------------------------------------------------
__device__ __forceinline__ float wave_sum_wmma(float p) {
  v2f a; a[0] = p;    a[1] = 0.0f;
  v2f b; b[0] = 1.0f; b[1] = 1.0f;
  v8f c = {};
  c = __builtin_amdgcn_wmma_f32_16x16x4_f32(
      /*neg_a=*/false, a, /*neg_b=*/false, b,
      /*c_mod=*/(short)0, c, /*reuse_a=*/false, /*reuse_b=*/false);
  float s = ((c[0] + c[1]) + (c[2] + c[3])) + ((c[4] + c[5]) + (c[6] + c[7]));
  s += __shfl_xor(s, 16, 32);   // combine lanes n and n+16 (wave32)
  return s;                     // every lane holds the wave total
}

// ---------------------------------------------------------------------------
// Kernel 2: weighted loss reduction (~10 MB read).
// total = sum_i [ (Wp*relu(Ax-b)^2 + Wd*relu(-lam)^2 + Wc*(lam*(Ax-b))^2)
//                 + Ws*(At_lam + c)^2 ] / (B*M)      (B*M == B*N)
// ---------------------------------------------------------------------------
__global__ void kkt_reduce(const float* __restrict__ ax_acc,
                           const float* __restrict__ at_acc,
                           const float* __restrict__ lam,
                           const float* __restrict__ b_pad,
                           const float* __restrict__ c_pad,
                           float* __restrict__ out) {
  int tid = blockIdx.x * blockDim.x + threadIdx.x;
  int stride = gridDim.x * blockDim.x;         // divides BM evenly -> no divergence
  const float inv = 1.0f / (float)BM;
  float acc = 0.0f;
  for (int i = tid; i < BM; i += stride) {
    float axmb = ax_acc[i] - b_pad[i];
    float l    = lam[i];
    float pr   = fmaxf(axmb, 0.0f);            // relu(Ax - b)
    float du   = fmaxf(-l, 0.0f);              // relu(-lam)
    float co   = l * axmb;                     // complementarity
    float st   = at_acc[i] + c_pad[i];         // stationarity
    acc += 0.1f * pr * pr + 0.1f * du * du + 0.2f * co * co + 0.6f * st * st;
  }
  acc *= inv;
  float wsum = wave_sum_wmma(acc);             // exact f32 wave reduction (v_wmma)
  if ((threadIdx.x & 31) == 0) atomicAdd(out, wsum);
}

// ---------------------------------------------------------------------------
// Launch. Inputs (setup_inputs order):
//   0: x_hat   [B*N] f32      1: lam_hat [B*M] f32
//   2: A_vals  [B,NNZ] f32    3: A_rows  [B,NNZ] i32   4: A_cols [B,NNZ] i32
//   5: b_pad   [B,M] f32      6: c_pad   [B,N] f32
//   7: b_mask (unused)        8: c_mask (unused)
// d_out: 1 float. d_ws: >= 4 MB (two B*M float accumulators).
// ---------------------------------------------------------------------------
extern "C" void kernel_launch(void* const* d_in, const int* in_sizes, int n_in,
                              void* d_out, int out_size, void* d_ws, size_t ws_size,
                              hipStream_t stream) {
  const float* x    = (const float*)d_in[0];
  const float* lam  = (const float*)d_in[1];
  const v4f*  vals  = (const v4f*)d_in[2];
  const v4i*  rows  = (const v4i*)d_in[3];
  const v4i*  cols  = (const v4i*)d_in[4];
  const float* bpad = (const float*)d_in[5];
  const float* cpad = (const float*)d_in[6];

  float* ax_acc = (float*)d_ws;               // B*M floats
  float* at_acc = ax_acc + BM;                // B*N floats
  float* out    = (float*)d_out;

  kkt_zero<<<1024, 256, 0, stream>>>(ax_acc, at_acc, out);
  // 524288 threads, 8 vec4 iterations each over the 192 MB COO stream.
  kkt_scatter<<<2048, 256, 0, stream>>>(x, lam, vals, rows, cols, ax_acc, at_acc);
  // 65536 threads, 8 iterations each; full waves -> EXEC all-ones for WMMA.
  kkt_reduce<<<256, 256, 0, stream>>>(ax_acc, at_acc, lam, bpad, cpad, out);
}